// Up_se_pam_61649960566931
// MI455X (gfx1250) — compile-verified
//
#include <hip/hip_runtime.h>
#include <math.h>
#include <stdint.h>

typedef __attribute__((ext_vector_type(2))) float v2f;
typedef __attribute__((ext_vector_type(8))) float v8f;

// Native fp32 WMMA: D(16x16) = A(16x4) * B(4x16) + C   (wave32, gfx1250)
__device__ __forceinline__ v8f wmma4(v2f a, v2f b, v8f c) {
  return __builtin_amdgcn_wmma_f32_16x16x4_f32(false, a, false, b, (short)0, c, false, false);
}

// ---- CDNA5 async global->LDS staging (ASYNCcnt path), with safe fallback ----
#if defined(__HIP_DEVICE_COMPILE__) && __has_builtin(__builtin_amdgcn_global_load_async_to_lds_b128)
#define USE_ASYNC_LDS 1
typedef int v4i __attribute__((ext_vector_type(4)));
typedef __attribute__((address_space(1))) v4i* gv4p;   // global AS pointer to 16B
typedef __attribute__((address_space(3))) float* lfp;  // LDS AS float pointer
typedef __attribute__((address_space(3))) v4i* lv4p;   // LDS AS pointer to 16B
__device__ __forceinline__ void cp16(float* dst_lds, const float* src_glb) {
  gv4p g = (gv4p)(uintptr_t)src_glb;    // AS1 addresses == flat values for global
  lfp  l0 = (lfp)dst_lds;               // addrspacecast generic -> LDS
  __builtin_amdgcn_global_load_async_to_lds_b128(g, (lv4p)l0, 0, 0);
}
__device__ __forceinline__ void async_join() {
#if __has_builtin(__builtin_amdgcn_s_wait_asynccnt)
  __builtin_amdgcn_s_wait_asynccnt(0);
#else
  asm volatile("s_wait_asynccnt 0" ::: "memory");
#endif
}
#else
#define USE_ASYNC_LDS 0
__device__ __forceinline__ void cp16(float* dst_lds, const float* src_glb) {
  *(float4*)dst_lds = *(const float4*)src_glb;
}
__device__ __forceinline__ void async_join() {}
#endif

// ---------------------------------------------------------------------------
// K0: weight transpose  [O][K] -> [K][O]  (makes WMMA A-fragment loads coalesced)
// ---------------------------------------------------------------------------
__global__ void k_transpose(const float* __restrict__ W, float* __restrict__ Wt,
                            int O, int K) {
  size_t i = (size_t)blockIdx.x * 256 + threadIdx.x;
  if (i < (size_t)O * K) {
    int o = (int)(i / K), k = (int)(i - (size_t)o * K);
    Wt[(size_t)k * O + o] = W[i];
  }
}

// ---------------------------------------------------------------------------
// K1: build concat[x2, up2x(x1)] into xcat and reduce per-(b,c) sums for SE.
// ---------------------------------------------------------------------------
__global__ void k_concat_se_reduce(const float* __restrict__ x1, const float* __restrict__ x2,
                                   float* __restrict__ xcat, float* __restrict__ sums) {
  int bc = blockIdx.x;           // 0..1023 : b*256 + c
  int b = bc >> 8, c = bc & 255;
  float acc = 0.f;
  const float scale = 31.f / 63.f;   // align_corners=True, 32 -> 64
  float* dst = xcat + ((size_t)bc << 12);
  if (c < 128) {
    const float* src = x2 + ((size_t)(b * 128 + c) << 12);
    for (int p = threadIdx.x; p < 4096; p += 256) {
      float v = src[p]; dst[p] = v; acc += v;
    }
  } else {
    const float* src = x1 + ((size_t)(b * 128 + (c - 128)) << 10);
    for (int p = threadIdx.x; p < 4096; p += 256) {
      int ho = p >> 6, wo = p & 63;
      float fy = ho * scale, fx = wo * scale;
      int y0 = (int)fy, x0 = (int)fx;
      int y1 = y0 + 1 > 31 ? 31 : y0 + 1;
      int x1i = x0 + 1 > 31 ? 31 : x0 + 1;
      float wy = fy - (float)y0, wx = fx - (float)x0;
      float v00 = src[y0 * 32 + x0], v01 = src[y0 * 32 + x1i];
      float v10 = src[y1 * 32 + x0], v11 = src[y1 * 32 + x1i];
      float v = (1.f - wy) * ((1.f - wx) * v00 + wx * v01)
              + wy * ((1.f - wx) * v10 + wx * v11);
      dst[p] = v; acc += v;
    }
  }
  __shared__ float red[256];
  red[threadIdx.x] = acc; __syncthreads();
  for (int s = 128; s > 0; s >>= 1) {
    if (threadIdx.x < s) red[threadIdx.x] += red[threadIdx.x + s];
    __syncthreads();
  }
  if (threadIdx.x == 0) sums[bc] = red[0];
}

// ---------------------------------------------------------------------------
// K2: SE MLP  s = sigmoid(relu(mean @ W1^T) @ W2^T)
// ---------------------------------------------------------------------------
__global__ void k_se_mlp(const float* __restrict__ sums, const float* __restrict__ w1,
                         const float* __restrict__ w2, float* __restrict__ svec) {
  __shared__ float hsh[64];
  int t = threadIdx.x;
  if (t < 64) {
    int b = t >> 4, j = t & 15;
    float h = 0.f;
    for (int c = 0; c < 256; ++c)
      h += (sums[b * 256 + c] * (1.f / 4096.f)) * w1[j * 256 + c];
    hsh[t] = fmaxf(h, 0.f);
  }
  __syncthreads();
  for (int idx = t; idx < 1024; idx += 256) {
    int b = idx >> 8, c = idx & 255;
    float v = 0.f;
#pragma unroll
    for (int j = 0; j < 16; ++j) v += hsh[b * 16 + j] * w2[c * 16 + j];
    svec[idx] = 1.f / (1.f + __expf(-v));
  }
}

// K3: x *= s (broadcast over HW; plane index = idx>>12)
__global__ void k_scale(float* __restrict__ xse, const float* __restrict__ svec) {
  size_t i = (size_t)blockIdx.x * 256 + threadIdx.x;
  xse[i] *= svec[i >> 12];
}

// ---------------------------------------------------------------------------
// K4: conv1x1 as GEMM via WMMA f32 16x16x4, transposed weights Wt[K][O].
// out[b][o][n] = sum_k Wt[k][o] * X[b][k][n] + bias[o];  K = 256, N = 4096.
// ---------------------------------------------------------------------------
__global__ void k_gemm1x1(const float* __restrict__ Wt, const float* __restrict__ bias,
                          const float* __restrict__ X, float* __restrict__ out, int O) {
  int lane = threadIdx.x & 31;
  int n0 = blockIdx.x * 16, m0 = blockIdx.y * 16, b = blockIdx.z;
  const float* Xb = X + ((size_t)b << 20);           // 256*4096
  float* outb = out + (size_t)b * O * 4096;
  int mrow = lane & 15, kgrp = lane >> 4;
  v8f acc = {};
  for (int k0 = 0; k0 < 256; k0 += 4) {
    int ka = k0 + 2 * kgrp;
    v2f a, bf;
    a.x = Wt[(size_t)ka * O + m0 + mrow];
    a.y = Wt[(size_t)(ka + 1) * O + m0 + mrow];
    bf.x = Xb[((size_t)ka << 12) + n0 + mrow];
    bf.y = Xb[((size_t)(ka + 1) << 12) + n0 + mrow];
    acc = wmma4(a, bf, acc);
  }
#pragma unroll
  for (int r = 0; r < 8; ++r) {
    int m = m0 + r + 8 * kgrp;
    outb[((size_t)m << 12) + n0 + mrow] = acc[r] + bias[m];
  }
}

// ---------------------------------------------------------------------------
// K5: PAM flash-attention.  Q = c_, K = b_ (Ci=32), V = d_ (C=256), N = 4096.
// 4 waves/block, each owning a 16-query tile; K/V j-tiles staged in LDS once
// per block via async global->LDS (ASYNCcnt), shared by all 4 waves.
// ---------------------------------------------------------------------------
#define VP 20   // LDS row pitch (floats): 16B-aligned rows, conflict-free reads

__global__ void k_attention(const float* __restrict__ bq, const float* __restrict__ cq,
                            const float* __restrict__ dv, const float* __restrict__ xse,
                            const float* __restrict__ gamma, float* __restrict__ y) {
  int wave = threadIdx.x >> 5, lane = threadIdx.x & 31;
  int b = blockIdx.y;
  int i0 = (blockIdx.x * 4 + wave) * 16;
  const float* Qp = cq + ((size_t)b * 32 << 12);
  const float* Kp = bq + ((size_t)b * 32 << 12);
  const float* Vp = dv + ((size_t)b * 256 << 12);
  int mrow = lane & 15, kgrp = lane >> 4;

  __shared__ float Kt[32 * VP];      // b_ tile  [k][jj]
  __shared__ float Vt[256 * VP];     // d_ tile  [c][jj]
  __shared__ float Pt[4][16][17];    // per-wave P tile for C->A relayout

  // Q fragments for this query tile (K=32 -> 8 fragments), registers, coalesced
  v2f qf[8];
#pragma unroll
  for (int kk = 0; kk < 8; ++kk) {
    int ka = kk * 4 + 2 * kgrp;
    qf[kk].x = Qp[((size_t)ka << 12) + i0 + mrow];
    qf[kk].y = Qp[((size_t)(ka + 1) << 12) + i0 + mrow];
  }

  float mrowv[8], lrowv[8];
  v8f ofrag[16];
#pragma unroll
  for (int r = 0; r < 8; ++r) { mrowv[r] = -1e30f; lrowv[r] = 0.f; }
#pragma unroll
  for (int cb = 0; cb < 16; ++cb) ofrag[cb] = (v8f){};

  for (int j0 = 0; j0 < 4096; j0 += 16) {
    __syncthreads();   // previous iteration's LDS consumers are done
    // ---- stage K tile (32x16) and V tile (256x16): 16B per transfer ----
    {
      int krow = threadIdx.x >> 2, seg = (threadIdx.x & 3) * 4;
      cp16(&Kt[krow * VP + seg], Kp + ((size_t)krow << 12) + j0 + seg);
#pragma unroll
      for (int it = 0; it < 8; ++it) {
        int i = threadIdx.x + (it << 7);
        int crow = i >> 2, vseg = (i & 3) * 4;
        cp16(&Vt[crow * VP + vseg], Vp + ((size_t)crow << 12) + j0 + vseg);
      }
    }
    async_join();
    __syncthreads();   // staged tiles visible to all waves

    // ---- S = Q K^T (16x16 tile, K=32) ----
    v8f s = {};
#pragma unroll
    for (int kk = 0; kk < 8; ++kk) {
      int ka = kk * 4 + 2 * kgrp;
      v2f kf;
      kf.x = Kt[ka * VP + mrow];
      kf.y = Kt[(ka + 1) * VP + mrow];
      s = wmma4(qf[kk], kf, s);
    }
    // ---- online softmax: slot r of this half-wave owns row (r + 8*kgrp) ----
    float alpha[8];
#pragma unroll
    for (int r = 0; r < 8; ++r) {
      float mx = s[r];
      mx = fmaxf(mx, __shfl_xor(mx, 1, 32));
      mx = fmaxf(mx, __shfl_xor(mx, 2, 32));
      mx = fmaxf(mx, __shfl_xor(mx, 4, 32));
      mx = fmaxf(mx, __shfl_xor(mx, 8, 32));
      float mn = fmaxf(mrowv[r], mx);
      alpha[r] = __expf(mrowv[r] - mn);
      mrowv[r] = mn;
      float p = __expf(s[r] - mn);
      s[r] = p;
      float rs = p;
      rs += __shfl_xor(rs, 1, 32);
      rs += __shfl_xor(rs, 2, 32);
      rs += __shfl_xor(rs, 4, 32);
      rs += __shfl_xor(rs, 8, 32);
      lrowv[r] = lrowv[r] * alpha[r] + rs;
    }
#pragma unroll
    for (int r = 0; r < 8; ++r) Pt[wave][r + 8 * kgrp][mrow] = s[r];
    __syncthreads();   // P tile visible (and own-wave DS ordered)

    // ---- O = O*alpha + P*V ----
#pragma unroll
    for (int cb = 0; cb < 16; ++cb)
#pragma unroll
      for (int r = 0; r < 8; ++r) ofrag[cb][r] *= alpha[r];
#pragma unroll
    for (int k0 = 0; k0 < 16; k0 += 4) {
      int ka = k0 + 2 * kgrp;
      v2f pa;
      pa.x = Pt[wave][mrow][ka];
      pa.y = Pt[wave][mrow][ka + 1];
#pragma unroll
      for (int cb = 0; cb < 16; ++cb) {
        v2f vf;
        vf.x = Vt[(cb * 16 + mrow) * VP + ka];
        vf.y = Vt[(cb * 16 + mrow) * VP + ka + 1];
        ofrag[cb] = wmma4(pa, vf, ofrag[cb]);
      }
    }
  }
  // epilogue: y[b][c][i] = gamma * pam + xse[b][c][i]
  float g = gamma[0];
#pragma unroll
  for (int cb = 0; cb < 16; ++cb)
#pragma unroll
    for (int r = 0; r < 8; ++r) {
      int m = r + 8 * kgrp;
      int c = cb * 16 + mrow;
      size_t idx = (((size_t)b * 256 + c) << 12) + i0 + m;
      y[idx] = g * (ofrag[cb][r] / lrowv[r]) + xse[idx];
    }
}

// ---------------------------------------------------------------------------
// K6: conv3x3 'SAME' as implicit GEMM via WMMA f32 16x16x4, Wt[K][O].
// ---------------------------------------------------------------------------
__device__ __forceinline__ float im2col(const float* __restrict__ Xb, int k, int h, int w) {
  int c = k / 9;
  int r = k - c * 9;
  int kh = r / 3;
  int kw = r - kh * 3;
  int hh = h + kh - 1, ww = w + kw - 1;
  if ((unsigned)hh >= 64u || (unsigned)ww >= 64u) return 0.f;
  return Xb[((size_t)c << 12) + hh * 64 + ww];
}

__global__ void k_conv3x3(const float* __restrict__ X, const float* __restrict__ Wt,
                          const float* __restrict__ bias, float* __restrict__ out,
                          int Cin, int O) {
  int lane = threadIdx.x & 31;
  int n0 = blockIdx.x * 16, m0 = blockIdx.y * 16, b = blockIdx.z;
  int K = Cin * 9;
  const float* Xb = X + ((size_t)b * Cin << 12);
  float* outb = out + (size_t)b * O * 4096;
  int mrow = lane & 15, kgrp = lane >> 4;
  int n = n0 + mrow, h = n >> 6, w = n & 63;
  v8f acc = {};
  for (int k0 = 0; k0 < K; k0 += 4) {
    int ka = k0 + 2 * kgrp;
    v2f a, bf;
    a.x = Wt[(size_t)ka * O + m0 + mrow];
    a.y = Wt[(size_t)(ka + 1) * O + m0 + mrow];
    bf.x = im2col(Xb, ka, h, w);
    bf.y = im2col(Xb, ka + 1, h, w);
    acc = wmma4(a, bf, acc);
  }
#pragma unroll
  for (int r = 0; r < 8; ++r) {
    int m = m0 + r + 8 * kgrp;
    outb[((size_t)m << 12) + n0 + mrow] = acc[r] + bias[m];
  }
}

// ---------------------------------------------------------------------------
// K7: GroupNorm stats (4 channels/group over HW): mean + rsqrt(var+eps)
// ---------------------------------------------------------------------------
__global__ void k_gn_stats(const float* __restrict__ X, float* __restrict__ stats, int C) {
  int g = blockIdx.x, b = blockIdx.y;
  const float* base = X + (((size_t)b * C + g * 4) << 12);
  float s = 0.f, s2 = 0.f;
  for (int i = threadIdx.x; i < 4 * 4096; i += 256) {
    float v = base[i]; s += v; s2 += v * v;
  }
  __shared__ float r1[256], r2[256];
  r1[threadIdx.x] = s; r2[threadIdx.x] = s2; __syncthreads();
  for (int st = 128; st > 0; st >>= 1) {
    if (threadIdx.x < st) { r1[threadIdx.x] += r1[threadIdx.x + st]; r2[threadIdx.x] += r2[threadIdx.x + st]; }
    __syncthreads();
  }
  if (threadIdx.x == 0) {
    float mu = r1[0] * (1.f / 16384.f);
    float var = r2[0] * (1.f / 16384.f) - mu * mu;
    stats[(b * 32 + g) * 2] = mu;
    stats[(b * 32 + g) * 2 + 1] = rsqrtf(var + 1e-5f);
  }
}

// K8: apply GN affine + LeakyReLU(0.2).  Total = 4*128*4096 elements.
__global__ void k_gn_apply(const float* __restrict__ X, const float* __restrict__ stats,
                           const float* __restrict__ gw, const float* __restrict__ gb,
                           float* __restrict__ out) {
  size_t i = (size_t)blockIdx.x * 256 + threadIdx.x;
  int c = (int)((i >> 12) & 127);
  int b = (int)(i >> 19);             // 128*4096 = 2^19
  int g = c >> 2;
  float mu = stats[(b * 32 + g) * 2];
  float rs = stats[(b * 32 + g) * 2 + 1];
  float v = (X[i] - mu) * rs * gw[c] + gb[c];
  out[i] = v >= 0.f ? v : 0.2f * v;
}

// ---------------------------------------------------------------------------
extern "C" void kernel_launch(void* const* d_in, const int* in_sizes, int n_in,
                              void* d_out, int out_size, void* d_ws, size_t ws_size,
                              hipStream_t stream) {
  const float* x1     = (const float*)d_in[0];
  const float* x2     = (const float*)d_in[1];
  const float* se_w1  = (const float*)d_in[2];
  const float* se_w2  = (const float*)d_in[3];
  const float* pam_wb = (const float*)d_in[4];
  const float* pam_bb = (const float*)d_in[5];
  const float* pam_wc = (const float*)d_in[6];
  const float* pam_bc = (const float*)d_in[7];
  const float* pam_wd = (const float*)d_in[8];
  const float* pam_bd = (const float*)d_in[9];
  const float* gamma  = (const float*)d_in[10];
  const float* dc_w1  = (const float*)d_in[11];
  const float* dc_b1  = (const float*)d_in[12];
  const float* gn1_w  = (const float*)d_in[13];
  const float* gn1_b  = (const float*)d_in[14];
  const float* dc_w2  = (const float*)d_in[15];
  const float* dc_b2  = (const float*)d_in[16];
  const float* gn2_w  = (const float*)d_in[17];
  const float* gn2_b  = (const float*)d_in[18];

  float* ws   = (float*)d_ws;
  float* xse  = ws + 0;           // 4*256*4096 = 4,194,304 (dead after attention)
  float* sums = ws + 4194304;     // 1024
  float* svec = ws + 4195328;     // 1024
  float* bq   = ws + 4196352;     // 524,288
  float* cq   = ws + 4720640;     // 524,288
  float* dv   = ws + 5244928;     // 4,194,304
  float* yb   = ws + 9439232;     // 4,194,304  (post-PAM x)
  float* c1   = ws + 13633536;    // 2,097,152
  float* g1   = ws + 15730688;    // 256
  float* a1   = ws + 15730944;    // 2,097,152
  float* wbT  = ws + 17828096;    // 8,192   (pam_wb^T  [256][32])
  float* wcT  = ws + 17836288;    // 8,192   (pam_wc^T)
  float* wdT  = ws + 17844480;    // 65,536  (pam_wd^T  [256][256])
  float* w1T  = ws + 17910016;    // 294,912 (dc_w1^T   [2304][128])
  float* w2T  = ws + 18204928;    // 147,456 (dc_w2^T   [1152][128])
  float* c2   = ws + 0;           // reuse dead xse region
  float* g2   = ws + 2097152;     // 256 (inside dead region)

  // weight transposes (independent of activations)
  k_transpose<<<32, 256, 0, stream>>>(pam_wb, wbT, 32, 256);
  k_transpose<<<32, 256, 0, stream>>>(pam_wc, wcT, 32, 256);
  k_transpose<<<256, 256, 0, stream>>>(pam_wd, wdT, 256, 256);
  k_transpose<<<1152, 256, 0, stream>>>(dc_w1, w1T, 128, 2304);
  k_transpose<<<576, 256, 0, stream>>>(dc_w2, w2T, 128, 1152);

  k_concat_se_reduce<<<1024, 256, 0, stream>>>(x1, x2, xse, sums);
  k_se_mlp<<<1, 256, 0, stream>>>(sums, se_w1, se_w2, svec);
  k_scale<<<16384, 256, 0, stream>>>(xse, svec);

  k_gemm1x1<<<dim3(256, 2, 4), 32, 0, stream>>>(wbT, pam_bb, xse, bq, 32);
  k_gemm1x1<<<dim3(256, 2, 4), 32, 0, stream>>>(wcT, pam_bc, xse, cq, 32);
  k_gemm1x1<<<dim3(256, 16, 4), 32, 0, stream>>>(wdT, pam_bd, xse, dv, 256);

  k_attention<<<dim3(64, 4), 128, 0, stream>>>(bq, cq, dv, xse, gamma, yb);

  k_conv3x3<<<dim3(256, 8, 4), 32, 0, stream>>>(yb, w1T, dc_b1, c1, 256, 128);
  k_gn_stats<<<dim3(32, 4), 256, 0, stream>>>(c1, g1, 128);
  k_gn_apply<<<8192, 256, 0, stream>>>(c1, g1, gn1_w, gn1_b, a1);

  k_conv3x3<<<dim3(256, 8, 4), 32, 0, stream>>>(a1, w2T, dc_b2, c2, 128, 128);
  k_gn_stats<<<dim3(32, 4), 256, 0, stream>>>(c2, g2, 128);
  k_gn_apply<<<8192, 256, 0, stream>>>(c2, g2, gn2_w, gn2_b, (float*)d_out);
}